// Synthesis_14929306320935
// MI455X (gfx1250) — compile-verified
//
#include <hip/hip_runtime.h>
#include <stdint.h>

// ---- problem constants (from reference) ----
#define N_ROWS  1048576
#define SWEEPS  32              // all 32 in the sweep loop (even -> result ends in smA)
#define TILE    512             // interior rows per block; 2048 blocks exactly
#define HALO    64              // 2 (radius) * 32 sweeps
#define EXT     (TILE + 2*HALO) // 640 resident rows per block
#define NT      256             // 8 wave32 per block
#define RCH     20              // rows per thread chunk: 32 chunks * 20 = EXT
#define LROW    12              // padded LDS row stride (floats): 48B, conflict-free + 16B aligned
#define LELEMS  ((EXT + 4) * LROW)  // +2 guard rows top & bottom; 30,912 B per buffer

#define NI_C    0.01f
#define BETA_C  0.5f
#define OMB_C   0.5f            // 1 - beta

// padded LDS index: local row r in [-2, EXT+1], column c in [0,8)
#define LIDX(r, c) ((((r) + 2) * LROW) + (c))

// ---- CDNA5 async global<->LDS path (guarded; fallback keeps compile clean) ----
#if defined(__has_builtin)
# if __has_builtin(__builtin_amdgcn_global_load_async_to_lds_b128) && \
     __has_builtin(__builtin_amdgcn_s_wait_asynccnt)
#  define HAVE_ASYNC_LD 1
# endif
# if __has_builtin(__builtin_amdgcn_global_store_async_from_lds_b128) && \
     __has_builtin(__builtin_amdgcn_s_wait_asynccnt)
#  define HAVE_ASYNC_ST 1
# endif
#endif

typedef int v4i __attribute__((ext_vector_type(4)));
typedef __attribute__((address_space(1))) v4i g_v4i;   // global int4
typedef __attribute__((address_space(3))) v4i l_v4i;   // LDS int4

__device__ __forceinline__ g_v4i* to_glb(const void* p) {
  return (g_v4i*)(uintptr_t)p;
}
__device__ __forceinline__ l_v4i* to_lds(const void* p) {
  // generic LDS pointer -> addrspace(3): low 32 bits are the LDS offset
  return (l_v4i*)(uint32_t)(uintptr_t)p;
}

extern "C" __global__ __launch_bounds__(NT)
void biharmonic_tblock_kernel(const float* __restrict__ y, float* __restrict__ out)
{
  __shared__ __align__(16) float smA[LELEMS];   // ping
  __shared__ __align__(16) float smB[LELEMS];   // pong

  const int  tid   = threadIdx.x;
  const int  c     = tid & 7;                   // owned column
  const int  rq    = (tid >> 3) * RCH;          // first owned local row
  const long g0    = (long)blockIdx.x * TILE;   // first interior row (global)
  const long gbase = g0 - HALO;                 // global row of local row 0
  const bool first = (g0 == 0);
  const bool last  = (g0 + TILE >= N_ROWS);

  // ---------- async fill smA with y over local rows [0, EXT) ----------
  {
    const float4* __restrict__ y4 = (const float4*)y;
    const long cbase = gbase * 2;               // first global float4 chunk
    for (int j = tid; j < EXT * 2; j += NT) {
      const long gc = cbase + j;
      if (gc >= 0 && gc < (long)N_ROWS * 2) {
        const int lo = (2 + (j >> 1)) * LROW + (j & 1) * 4;
#if HAVE_ASYNC_LD
        __builtin_amdgcn_global_load_async_to_lds_b128(
            to_glb(&y4[gc]), to_lds(&smA[lo]), 0, 0);
#else
        *(float4*)&smA[lo] = y4[gc];
#endif
      }
    }
  }

  // ---------- y for owned rows lives in registers (never changes) ----------
  float yreg[RCH];
  #pragma unroll
  for (int k = 0; k < RCH; ++k) {
    const long g = gbase + rq + k;
    yreg[k] = (g >= 0 && g < N_ROWS) ? y[g * 8 + c] : 0.0f;
  }

#if HAVE_ASYNC_LD
  __builtin_amdgcn_s_wait_asynccnt(0);
#endif
  __syncthreads();

  // ---------- 32 Jacobi sweeps in LDS, register sliding window ----------
  float* cur = smA;
  float* nxt = smB;

  if (!first && !last) {
    // 2046 of 2048 blocks: every computed row is strictly interior (2<=g<=N-3)
    for (int s = 1; s <= SWEEPS; ++s) {
      const int r_lo = 2 * s;
      const int r_hi = EXT - 2 * s;
      float a  = cur[LIDX(rq - 2, c)];
      float b  = cur[LIDX(rq - 1, c)];
      float c0 = cur[LIDX(rq,     c)];
      float d  = cur[LIDX(rq + 1, c)];
      #pragma unroll
      for (int k = 0; k < RCH; ++k) {
        const int   r  = rq + k;
        const float e5 = cur[LIDX(r + 2, c)];
        const float lap = a - 4.0f * b + 6.0f * c0 - 4.0f * d + e5;
        const float o   = c0 - NI_C * (OMB_C * lap + BETA_C * (c0 - yreg[k]));
        if (r >= r_lo && r < r_hi) nxt[LIDX(r, c)] = o;
        a = b; b = c0; c0 = d; d = e5;
      }
      __syncthreads();
      float* t = cur; cur = nxt; nxt = t;
    }
  } else {
    // boundary blocks: rows 0/1 and N-2/N-1 use closed-form updates, so the
    // physical edge stays exact and never shrinks
    for (int s = 1; s <= SWEEPS; ++s) {
      const int r_lo = first ? HALO : 2 * s;
      const int r_hi = last  ? (EXT - HALO) : (EXT - 2 * s);
      float a  = cur[LIDX(rq - 2, c)];
      float b  = cur[LIDX(rq - 1, c)];
      float c0 = cur[LIDX(rq,     c)];
      float d  = cur[LIDX(rq + 1, c)];
      #pragma unroll
      for (int k = 0; k < RCH; ++k) {
        const int   r  = rq + k;
        const float e5 = cur[LIDX(r + 2, c)];
        const long  g  = gbase + r;
        float o;
        if (g >= 2 && g < N_ROWS - 2) {
          const float lap = a - 4.0f * b + 6.0f * c0 - 4.0f * d + e5;
          o = c0 - NI_C * (OMB_C * lap + BETA_C * (c0 - yreg[k]));
        } else if (g >= N_ROWS - 2) {
          o = yreg[k];                       // last two rows reset to y
        } else if (g == 1) {                 // one-sided: rows g=0..3 are b,c0,d,e5
          const float l1 = -2.0f * b + 5.0f * c0 - 4.0f * d + e5;
          o = c0 - NI_C * (OMB_C * l1 + BETA_C * (c0 - yreg[k]));
        } else {                             // g == 0: copy old value
          o = c0;
        }
        if (r >= r_lo && r < r_hi) nxt[LIDX(r, c)] = o;
        a = b; b = c0; c0 = d; d = e5;
      }
      __syncthreads();
      float* t = cur; cur = nxt; nxt = t;
    }
  }
  // SWEEPS even -> final result sits in `cur` (== smA)

  // ---------- async write back interior rows [g0, g0+TILE) ----------
  {
    float4* __restrict__ out4 = (float4*)out;
    for (int j = tid; j < TILE * 2; j += NT) {
      const int  lo = (2 + HALO + (j >> 1)) * LROW + (j & 1) * 4;
      const long gc = g0 * 2 + j;
#if HAVE_ASYNC_ST
      __builtin_amdgcn_global_store_async_from_lds_b128(
          to_glb(&out4[gc]), to_lds(&cur[lo]), 0, 0);
#else
      out4[gc] = *(float4*)&cur[lo];
#endif
    }
#if HAVE_ASYNC_ST
    __builtin_amdgcn_s_wait_asynccnt(0);
#endif
  }
}

extern "C" void kernel_launch(void* const* d_in, const int* in_sizes, int n_in,
                              void* d_out, int out_size, void* d_ws, size_t ws_size,
                              hipStream_t stream) {
  (void)in_sizes; (void)n_in; (void)d_ws; (void)ws_size; (void)out_size;
  const float* y = (const float*)d_in[0];
  float* out = (float*)d_out;
  hipLaunchKernelGGL(biharmonic_tblock_kernel,
                     dim3(N_ROWS / TILE), dim3(NT), 0, stream, y, out);
}